// WassersteinLogGS_11897059410347
// MI455X (gfx1250) — compile-verified
//
#include <hip/hip_runtime.h>
#include <stdint.h>

#define DEVINL __device__ __forceinline__

static constexpr float kEPS = 1e-8f;
static constexpr float kTwoPiOver3 = 2.0943951023931953f;

// --- CDNA5 async Global<->LDS (ASYNCcnt) helpers ---------------------------
// GVS addressing: mem = SGPR base + 32-bit VGPR offset. LDS address comes from
// a VGPR holding the LDS byte offset (low 32 bits of the generic pointer).
DEVINL void async_load_b128(uint32_t lds_addr, uint32_t voff, const void* sbase) {
  asm volatile("global_load_async_to_lds_b128 %0, %1, %2"
               :
               : "v"(lds_addr), "v"(voff), "s"(sbase)
               : "memory");
}
DEVINL void async_store_b128(uint32_t lds_addr, uint32_t voff, void* sbase) {
  asm volatile("global_store_async_from_lds_b128 %0, %1, %2"
               :
               : "v"(voff), "v"(lds_addr), "s"(sbase)
               : "memory");
}
DEVINL void wait_asynccnt0() {
  asm volatile("s_wait_asynccnt 0x0" ::: "memory");
}
DEVINL uint32_t lds_offset_of(const void* generic_ptr) {
  return (uint32_t)(uintptr_t)generic_ptr;  // low 32 bits = LDS byte address
}

DEVINL float fast_rcp(float x) { return __builtin_amdgcn_rcpf(x); }

// --- Eigenvalues of a symmetric 3x3 (trigonometric / Cardano method) -------
DEVINL void eig3_sym(const float M[3][3], float& e1, float& e2, float& e3) {
  const float a01 = M[0][1], a02 = M[0][2], a12 = M[1][2];
  const float q  = (M[0][0] + M[1][1] + M[2][2]) * (1.0f / 3.0f);
  const float b0 = M[0][0] - q, b1 = M[1][1] - q, b2 = M[2][2] - q;
  const float p1 = a01 * a01 + a02 * a02 + a12 * a12;
  const float p2 = b0 * b0 + b1 * b1 + b2 * b2 + 2.0f * p1;
  const float p  = sqrtf(p2 * (1.0f / 6.0f));
  const float pin = (p > 1e-24f) ? fast_rcp(p) : 0.0f;
  const float c0 = b0 * pin, c1 = b1 * pin, c2 = b2 * pin;
  const float d01 = a01 * pin, d02 = a02 * pin, d12 = a12 * pin;
  float det = c0 * (c1 * c2 - d12 * d12)
            - d01 * (d01 * c2 - d12 * d02)
            + d02 * (d01 * d12 - c1 * d02);
  float r = 0.5f * det;
  r = fminf(1.0f, fmaxf(-1.0f, r));
  const float phi = acosf(r) * (1.0f / 3.0f);
  const float tp = 2.0f * p;
  e1 = q + tp * __cosf(phi);                 // max
  e3 = q + tp * __cosf(phi + kTwoPiOver3);   // min
  e2 = 3.0f * q - e1 - e3;
}

// Guarded divided difference; 0 on coincident nodes is exact for
// value-interpolation on the spectrum of a diagonalizable matrix.
DEVINL float dd(float fa, float fb, float a, float b) {
  const float d = a - b;
  const float v = (fa - fb) * fast_rcp(d);
  return (fabsf(d) > 1e-24f) ? v : 0.0f;
}

// p(x) = c0 + c1 x + c2 x^2 interpolating (l_i, f_i); then p(A) = f(A).
DEVINL void fpoly(float l1, float l2, float l3, float f1, float f2, float f3,
                  float& c0, float& c1, float& c2) {
  const float d12  = dd(f1, f2, l1, l2);
  const float d23  = dd(f2, f3, l2, l3);
  const float d123 = dd(d12, d23, l1, l3);
  c2 = d123;
  c1 = d12 - d123 * (l1 + l2);
  c0 = f1 - d12 * l1 + d123 * (l1 * l2);
}

DEVINL void mm3(const float A[3][3], const float B[3][3], float C[3][3]) {
#pragma unroll
  for (int i = 0; i < 3; ++i)
#pragma unroll
    for (int j = 0; j < 3; ++j) {
      float acc = A[i][0] * B[0][j];
      acc = fmaf(A[i][1], B[1][j], acc);
      acc = fmaf(A[i][2], B[2][j], acc);
      C[i][j] = acc;
    }
}

DEVINL void polymat(const float A[3][3], const float A2[3][3],
                    float c0, float c1, float c2, float R[3][3]) {
#pragma unroll
  for (int i = 0; i < 3; ++i)
#pragma unroll
    for (int j = 0; j < 3; ++j) {
      float v = fmaf(c1, A[i][j], c2 * A2[i][j]);
      if (i == j) v += c0;
      R[i][j] = v;
    }
}

__global__ __launch_bounds__(256) void WassersteinLogGS_kernel(
    const float* __restrict__ miu1, const float* __restrict__ miu2,
    const float* __restrict__ cov1, const float* __restrict__ cov2,
    float* __restrict__ out_mu, float* __restrict__ out_cov, int B) {
  __shared__ __align__(16) float s1buf[256 * 9];
  __shared__ __align__(16) float s2buf[256 * 9];

  const int t     = threadIdx.x;
  const int ebase = blockIdx.x * 256;     // first element of this tile
  const int gbase = ebase * 9;            // flat float index into cov tensors
  const int gend  = 9 * B;
  const bool full = (ebase + 256 <= B);   // uniform per block

  // Prefetch next tile into cache (gfx1250 global_prefetch_b8); speculative.
  {
    const int pf = gbase + 9 * 256 + t * 9;
    if (pf < gend) {
      __builtin_prefetch(cov1 + pf, 0, 1);
      __builtin_prefetch(cov2 + pf, 0, 1);
    }
  }

  if (full) {
    // CDNA5 async copy, 16 B/lane: 9216 B/buffer = 2 full rounds + 1 quarter
    // round (EXEC-masked). No VGPR round-trip.
    const uint32_t l1 = lds_offset_of(&s1buf[0]);
    const uint32_t l2 = lds_offset_of(&s2buf[0]);
    const float* p1 = cov1 + gbase;
    const float* p2 = cov2 + gbase;
    const uint32_t o0 = (uint32_t)t * 16u;
    async_load_b128(l1 + o0, o0, p1);
    async_load_b128(l2 + o0, o0, p2);
    const uint32_t o1 = o0 + 4096u;
    async_load_b128(l1 + o1, o1, p1);
    async_load_b128(l2 + o1, o1, p2);
    if (t < 64) {
      const uint32_t o2 = o0 + 8192u;
      async_load_b128(l1 + o2, o2, p1);
      async_load_b128(l2 + o2, o2, p2);
    }
    wait_asynccnt0();  // this wave's LDS writes landed before the barrier
  } else {
    for (int k = 0; k < 9; ++k) {
      const int idx = gbase + t + k * 256;
      const bool ok = (idx < gend);
      s1buf[t + k * 256] = ok ? cov1[idx] : 0.0f;
      s2buf[t + k * 256] = ok ? cov2[idx] : 0.0f;
    }
  }
  __syncthreads();

  // miu2 - miu1: one float4 round covers the tile's 3072-B span.
  {
    const int mbase = ebase * 3;
    if (full) {
      if (t < 192) {
        const float4* a = (const float4*)(miu1 + mbase);
        const float4* b = (const float4*)(miu2 + mbase);
        float4* o = (float4*)(out_mu + mbase);
        const float4 x = a[t];
        const float4 y = b[t];
        float4 r;
        r.x = y.x - x.x; r.y = y.y - x.y; r.z = y.z - x.z; r.w = y.w - x.w;
        o[t] = r;
      }
    } else {
      const int mend = 3 * B;
      for (int k = 0; k < 3; ++k) {
        const int idx = mbase + t + k * 256;
        if (idx < mend) out_mu[idx] = miu2[idx] - miu1[idx];
      }
    }
  }

  // ---- Per-element computation (bank-conflict-free: gcd(9,64)=1) ----------
  float S1[3][3], A[3][3], S2s[3][3];
#pragma unroll
  for (int i = 0; i < 3; ++i)
#pragma unroll
    for (int j = 0; j < 3; ++j) S1[i][j] = s1buf[9 * t + 3 * i + j];
#pragma unroll
  for (int i = 0; i < 3; ++i)
#pragma unroll
    for (int j = 0; j < 3; ++j) {
      A[i][j]   = 0.5f * (S1[i][j] + S1[j][i]);
      S2s[i][j] = 0.5f * (s2buf[9 * t + 3 * i + j] + s2buf[9 * t + 3 * j + i]);
    }

  // eigh(sym(cov1)); clipped sqrt and inverse-sqrt as matrix polynomials.
  float l1, l2, l3;
  eig3_sym(A, l1, l2, l3);
  const float h1 = sqrtf(fmaxf(l1, kEPS));
  const float h2 = sqrtf(fmaxf(l2, kEPS));
  const float h3 = sqrtf(fmaxf(l3, kEPS));
  const float g1 = fast_rcp(kEPS + h1);
  const float g2 = fast_rcp(kEPS + h2);
  const float g3 = fast_rcp(kEPS + h3);

  float fa0, fa1, fa2, fb0, fb1, fb2;
  fpoly(l1, l2, l3, h1, h2, h3, fa0, fa1, fa2);
  fpoly(l1, l2, l3, g1, g2, g3, fb0, fb1, fb2);

  float A2[3][3];
  mm3(A, A, A2);
  float Asq[3][3], Ais[3][3];
  polymat(A, A2, fa0, fa1, fa2, Asq);   // cov1^{1/2}
  polymat(A, A2, fb0, fb1, fb2, Ais);   // cov1^{-1/2} (eps-regularized)

  // C = sym(Asq @ cov2 @ Asq^T) + eps*I  (Asq symmetric => = Asq sym(cov2) Asq)
  float T[3][3], Cf[3][3], C[3][3];
  mm3(Asq, S2s, T);
  mm3(T, Asq, Cf);
#pragma unroll
  for (int i = 0; i < 3; ++i)
#pragma unroll
    for (int j = 0; j < 3; ++j) {
      float v = 0.5f * (Cf[i][j] + Cf[j][i]);
      if (i == j) v += kEPS;
      C[i][j] = v;
    }

  // sqrt of Hermitian: Q diag(sqrt|mu|) Q^T  == polynomial in C.
  float m1, m2, m3;
  eig3_sym(C, m1, m2, m3);
  const float k1 = sqrtf(fabsf(m1));
  const float k2 = sqrtf(fabsf(m2));
  const float k3 = sqrtf(fabsf(m3));
  float gc0, gc1, gc2;
  fpoly(m1, m2, m3, k1, k2, k3, gc0, gc1, gc2);

  float C2[3][3], Csq[3][3];
  mm3(C, C, C2);
  polymat(C, C2, gc0, gc1, gc2, Csq);

  // V = Asq @ Csq @ Ais ; out = V + V^T - cov1 - cov1^T
  float W[3][3], V[3][3];
  mm3(Asq, Csq, W);
  mm3(W, Ais, V);

  // Stage result in LDS (each thread owns exactly its 9-float region).
#pragma unroll
  for (int i = 0; i < 3; ++i)
#pragma unroll
    for (int j = 0; j < 3; ++j)
      s1buf[9 * t + 3 * i + j] = V[i][j] + V[j][i] - S1[i][j] - S1[j][i];
  __syncthreads();

  // Coalesced writeback of the 3x3 outputs.
  if (full) {
    // CDNA5 async copy LDS -> global, 16 B/lane.
    const uint32_t l1 = lds_offset_of(&s1buf[0]);
    float* po = out_cov + gbase;
    const uint32_t o0 = (uint32_t)t * 16u;
    async_store_b128(l1 + o0, o0, po);
    const uint32_t o1 = o0 + 4096u;
    async_store_b128(l1 + o1, o1, po);
    if (t < 64) {
      const uint32_t o2 = o0 + 8192u;
      async_store_b128(l1 + o2, o2, po);
    }
    wait_asynccnt0();
  } else {
    for (int k = 0; k < 9; ++k) {
      const int idx = gbase + t + k * 256;
      if (idx < gend) out_cov[idx] = s1buf[t + k * 256];
    }
  }
}

extern "C" void kernel_launch(void* const* d_in, const int* in_sizes, int n_in,
                              void* d_out, int out_size, void* d_ws, size_t ws_size,
                              hipStream_t stream) {
  const float* miu1 = (const float*)d_in[0];
  const float* miu2 = (const float*)d_in[1];
  const float* cov1 = (const float*)d_in[2];
  const float* cov2 = (const float*)d_in[3];
  const int B = in_sizes[0] / 3;

  float* out_mu  = (float*)d_out;
  float* out_cov = out_mu + (size_t)3 * B;

  const int blocks = (B + 255) / 256;
  WassersteinLogGS_kernel<<<blocks, 256, 0, stream>>>(miu1, miu2, cov1, cov2,
                                                      out_mu, out_cov, B);
}